// OrthogonalityRegularization_18674517803011
// MI455X (gfx1250) — compile-verified
//
#include <hip/hip_runtime.h>
#include <hip/hip_bf16.h>

typedef __attribute__((ext_vector_type(2))) float v2f;
typedef __attribute__((ext_vector_type(8))) float v8f;

#define ROWS 4096
#define COLS 3072
#define CT   (COLS / 16)     // 192 column tiles
#define RS   8               // row slabs -> 1536 blocks
#define WAVES_PER_BLOCK 8    // 256 threads = 8 wave32
#define ROWS_PER_WAVE (ROWS / (RS * WAVES_PER_BLOCK))   // 64

// Phase 1: per-column sums of x^2 via V_WMMA_F32_16X16X4_F32 (A = ones),
// plus sum of x^4 on the VALU while the data is in registers.
__global__ void __launch_bounds__(256)
orth_reg_phase1(const float* __restrict__ x,
                float* __restrict__ colsum,   // [COLS]
                float* __restrict__ frob)     // [1]
{
    __shared__ float colacc[16];
    __shared__ float frred[WAVES_PER_BLOCK];

    const int lane = threadIdx.x & 31;
    const int wave = threadIdx.x >> 5;             // 0..7
    const int half = lane >> 4;                    // 0 or 1
    const int ct   = blockIdx.x % CT;              // adjacent blocks -> adjacent 64B
    const int rs   = blockIdx.x / CT;              //   segments of the same rows
    const int col  = ct * 16 + (lane & 15);

    if (threadIdx.x < 16) colacc[threadIdx.x] = 0.0f;
    __syncthreads();

    const int r0 = rs * (ROWS / RS) + wave * ROWS_PER_WAVE;

    v8f acc = {};
    v2f a;  a.x = 1.0f; a.y = 1.0f;                // ones A-matrix (layout-invariant)
    float fr = 0.0f;

    const float* __restrict__ xc = x + (size_t)r0 * COLS + col;

#pragma unroll 4
    for (int k = 0; k < ROWS_PER_WAVE; k += 4) {
        // B tile: 4 rows x 16 cols of Y = x^2.  Any K<->(vgpr,half) permutation
        // is fine under an all-ones A, so each half-wave takes two rows.
        float x0 = xc[(size_t)(k + half) * COLS];
        float x1 = xc[(size_t)(k + 2 + half) * COLS];
        float y0 = x0 * x0;
        float y1 = x1 * x1;
        fr = fmaf(y0, y0, fr);                     // x^4 accumulation (Frobenius)
        fr = fmaf(y1, y1, fr);
        v2f b;  b.x = y0;  b.y = y1;
        // C += ones(16x4) * B(4x16): every C row becomes the running column sums.
        acc = __builtin_amdgcn_wmma_f32_16x16x4_f32(
                  /*neg_a=*/false, a, /*neg_b=*/false, b,
                  /*c_mod=*/(short)0, acc, /*reuse_a=*/false, /*reuse_b=*/false);
    }

    // C/D layout: VGPR0 lanes 0-15 hold (M=0, N=0..15); lanes 16-31 duplicate
    // it at M=8 — take only the first half-wave, fold into LDS (ds_add_f32).
    if (lane < 16)
        atomicAdd(&colacc[lane], acc[0]);

    // wave32 reduction of the x^4 partial, one LDS slot per wave.
    for (int off = 16; off > 0; off >>= 1)
        fr += __shfl_xor(fr, off, 32);
    if (lane == 0) frred[wave] = fr;
    __syncthreads();

    // One global atomic per column per block + one frob atomic per block.
    if (wave == 0) {
        if (lane < 16)
            atomicAdd(&colsum[col], colacc[lane]);
        if (lane == 0) {
            float t = 0.0f;
#pragma unroll
            for (int w = 0; w < WAVES_PER_BLOCK; ++w) t += frred[w];
            atomicAdd(frob, t);
        }
    }
}

// Phase 2: out = ( sum_d colsum[d]^2 - frob ) / D
__global__ void __launch_bounds__(256)
orth_reg_phase2(const float* __restrict__ ws,   // [COLS] colsum, ws[COLS] = frob
                float* __restrict__ out)
{
    __shared__ float red[WAVES_PER_BLOCK];
    const int lane = threadIdx.x & 31;
    const int wave = threadIdx.x >> 5;

    float s = 0.0f;
    for (int i = threadIdx.x; i < COLS; i += blockDim.x) {
        float v = ws[i];
        s = fmaf(v, v, s);
    }
    for (int off = 16; off > 0; off >>= 1)
        s += __shfl_xor(s, off, 32);
    if (lane == 0) red[wave] = s;
    __syncthreads();

    if (wave == 0) {
        float t = (lane < WAVES_PER_BLOCK) ? red[lane] : 0.0f;
        for (int off = 16; off > 0; off >>= 1)
            t += __shfl_xor(t, off, 32);
        if (lane == 0)
            out[0] = (t - ws[COLS]) * (1.0f / (float)COLS);
    }
}

extern "C" void kernel_launch(void* const* d_in, const int* in_sizes, int n_in,
                              void* d_out, int out_size, void* d_ws, size_t ws_size,
                              hipStream_t stream) {
    (void)in_sizes; (void)n_in; (void)out_size; (void)ws_size;

    const float* x = (const float*)d_in[0];       // [4096, 3072] f32
    float* ws      = (float*)d_ws;                // [COLS] colsum + [1] frob
    float* out     = (float*)d_out;               // scalar f32

    // Zero the accumulators (graph-capture-safe stream memset).
    hipMemsetAsync(d_ws, 0, (COLS + 1) * sizeof(float), stream);

    // 192 column tiles x 8 row slabs = 1536 blocks; 12288 waves streaming.
    orth_reg_phase1<<<CT * RS, 32 * WAVES_PER_BLOCK, 0, stream>>>(x, ws, ws + COLS);
    orth_reg_phase2<<<1, 256, 0, stream>>>(ws, out);
}